// MLPGCN_78804059947205
// MI455X (gfx1250) — compile-verified
//
#include <hip/hip_runtime.h>
#include <math.h>

typedef __attribute__((ext_vector_type(2))) float v2f;
typedef __attribute__((ext_vector_type(8))) float v8f;

#define IN_F   64
#define HID_F  64
#define HID_CH 32
#define OUT_F  16

static __device__ __forceinline__ v8f zero8() {
  v8f z;
#pragma unroll
  for (int i = 0; i < 8; ++i) z[i] = 0.f;
  return z;
}

// ---------------------------------------------------------------- init ------
__global__ void init_ws_k(float* p, long n) {
  long i = (long)blockIdx.x * blockDim.x + threadIdx.x;
  long stride = (long)gridDim.x * blockDim.x;
  for (; i < n; i += stride) p[i] = 0.f;
}

// ------------------------------------------------------------- degrees ------
__global__ void degree_k(const int* __restrict__ src, const int* __restrict__ dst,
                         float* deg_out, float* deg_in, int E) {
  int e = blockIdx.x * 256 + threadIdx.x;
  if (e < E) {
    atomicAdd(&deg_out[src[e]], 1.f);
    atomicAdd(&deg_in[dst[e]], 1.f);
  }
}

__global__ void invsqrt_k(float* d, int n) {
  int i = blockIdx.x * 256 + threadIdx.x;
  if (i < n) d[i] = rsqrtf(fmaxf(d[i], 1.f));
}

// ---------------------------------------------- h1 / x dual GEMM (WMMA) -----
// h1 = tanh(feat @ fc1_w^T + fc1_b) ; x = feat @ filt_w^T + filt_b
// block: 256 threads = 8 waves, 16 rows/wave -> 128 rows/block
__global__ __launch_bounds__(256) void gemm_h1x_k(
    const float* __restrict__ feat,
    const float* __restrict__ w1, const float* __restrict__ b1,
    const float* __restrict__ w2, const float* __restrict__ b2,
    float* __restrict__ h1, float* __restrict__ xb, int N) {
  __shared__ float sF[128 * 64];   // feat tile
  __shared__ float sW1[64 * 64];   // fc1_w  (row j, col k)
  __shared__ float sW2[64 * 64];   // filt_w
  const int tid = threadIdx.x;
  const int rowbase = blockIdx.x * 128;

  for (int i = tid; i < 128 * 64; i += 256) {
    int r = i >> 6, c = i & 63;
    int g = rowbase + r;
    sF[i] = (g < N) ? feat[g * 64 + c] : 0.f;
  }
  for (int i = tid; i < 64 * 64; i += 256) { sW1[i] = w1[i]; sW2[i] = w2[i]; }
  __syncthreads();

  const int wave = tid >> 5, lane = tid & 31;
  const int m = lane & 15, hi = lane >> 4;   // A: row m / K-half; B,D: col n=m / M-half
  const int lrow = wave * 16 + m;

  v8f acc1[4], acc2[4];
#pragma unroll
  for (int nt = 0; nt < 4; ++nt) { acc1[nt] = zero8(); acc2[nt] = zero8(); }

#pragma unroll
  for (int ks = 0; ks < 16; ++ks) {           // K = 64 in steps of 4
    const int kb = 4 * ks + 2 * hi;
    v2f a = *(const v2f*)&sF[lrow * 64 + kb];
#pragma unroll
    for (int nt = 0; nt < 4; ++nt) {
      const int j = nt * 16 + m;              // output column = weight row
      v2f bA = *(const v2f*)&sW1[j * 64 + kb];
      v2f bB = *(const v2f*)&sW2[j * 64 + kb];
      acc1[nt] = __builtin_amdgcn_wmma_f32_16x16x4_f32(false, a, false, bA,
                                                       (short)0, acc1[nt], false, false);
      acc2[nt] = __builtin_amdgcn_wmma_f32_16x16x4_f32(false, a, false, bB,
                                                       (short)0, acc2[nt], false, false);
    }
  }

  const int gbase = rowbase + wave * 16 + hi * 8;
#pragma unroll
  for (int nt = 0; nt < 4; ++nt) {
    const int col = nt * 16 + m;
    const float bb1 = b1[col], bb2 = b2[col];
#pragma unroll
    for (int r = 0; r < 8; ++r) {
      int g = gbase + r;
      if (g < N) {
        h1[g * 64 + col] = tanhf(acc1[nt][r] + bb1);
        xb[g * 64 + col] = acc2[nt][r] + bb2;
      }
    }
  }
}

// --------------------------------------------------- edge scatter-add -------
__global__ void scatter_k(const int* __restrict__ src, const int* __restrict__ dst,
                          const float* __restrict__ inv_out, const float* __restrict__ inv_in,
                          const float* __restrict__ xb, float* __restrict__ h2, int E) {
  int t = blockIdx.x * 256 + threadIdx.x;
  int e = t >> 6;
  int k = t & 63;
  if (e < E) {
    int s = src[e], d = dst[e];
    float coef = inv_out[s] * inv_in[d];
    atomicAdd(&h2[d * 64 + k], xb[s * 64 + k] * coef);
  }
}

// ------------------------------------------- channel attention sums ---------
// wave per (node, channel) pair; lane = hidden channel of att MLP
__global__ __launch_bounds__(256) void att_k(
    const float* __restrict__ h1, const float* __restrict__ h2,
    const float* __restrict__ w1, const float* __restrict__ bb1,
    const float* __restrict__ w2v, float* sumW, int N) {
  __shared__ float sW1[64 * 32];
  __shared__ float sPart[8];
  const int tid = threadIdx.x;
  for (int i = tid; i < 64 * 32; i += 256) sW1[i] = w1[i];
  __syncthreads();

  const int wave = tid >> 5, lane = tid & 31;
  const long p = (long)blockIdx.x * 8 + wave;
  float val = 0.f;
  if (p < 2L * N) {
    const int node = (int)(p >> 1);
    const int c = (int)(p & 1);
    const float* z = (c == 0 ? h1 : h2) + (long)node * 64;
    float s = 0.f;
#pragma unroll
    for (int k = 0; k < 64; ++k) s += z[k] * sW1[k * 32 + lane];
    val = tanhf(s + bb1[lane]) * w2v[lane];
  }
  for (int off = 16; off > 0; off >>= 1) val += __shfl_down(val, off, 32);
  if (lane == 0) sPart[wave] = val;
  __syncthreads();
  if (tid < 2) {  // wave parity == channel (block base is a multiple of 8)
    float s = sPart[tid] + sPart[tid + 2] + sPart[tid + 4] + sPart[tid + 6];
    atomicAdd(&sumW[tid], s);
  }
}

__global__ void beta_k(const float* sumW, float* beta, int N) {
  if (threadIdx.x == 0) {
    float m0 = sumW[0] / (float)N, m1 = sumW[1] / (float)N;
    float mx = fmaxf(m0, m1);
    float e0 = __expf(m0 - mx), e1 = __expf(m1 - mx);
    float inv = 1.f / (e0 + e1);
    beta[0] = e0 * inv;
    beta[1] = e1 * inv;
  }
}

// ----------------------------------- fused output GEMM (WMMA, OUT_F=16) -----
__global__ __launch_bounds__(256) void out_k(
    const float* __restrict__ h1, const float* __restrict__ h2,
    const float* __restrict__ beta,
    const float* __restrict__ w, const float* __restrict__ b,
    float* __restrict__ out, int N) {
  __shared__ float sF[128 * 64];
  __shared__ float sW[16 * 64];
  const int tid = threadIdx.x;
  const int rowbase = blockIdx.x * 128;
  const float b0 = beta[0], b1v = beta[1];

  for (int i = tid; i < 128 * 64; i += 256) {
    int r = i >> 6, c = i & 63;
    int g = rowbase + r;
    sF[i] = (g < N) ? (b0 * h1[g * 64 + c] + b1v * h2[g * 64 + c]) : 0.f;
  }
  for (int i = tid; i < 16 * 64; i += 256) sW[i] = w[i];
  __syncthreads();

  const int wave = tid >> 5, lane = tid & 31;
  const int m = lane & 15, hi = lane >> 4;
  const int lrow = wave * 16 + m;

  v8f acc = zero8();
#pragma unroll
  for (int ks = 0; ks < 16; ++ks) {
    const int kb = 4 * ks + 2 * hi;
    v2f a = *(const v2f*)&sF[lrow * 64 + kb];
    v2f bf = *(const v2f*)&sW[m * 64 + kb];   // fc2_w row = output col
    acc = __builtin_amdgcn_wmma_f32_16x16x4_f32(false, a, false, bf,
                                                (short)0, acc, false, false);
  }
  const int gbase = rowbase + wave * 16 + hi * 8;
  const float bias = b[m];
#pragma unroll
  for (int r = 0; r < 8; ++r) {
    int g = gbase + r;
    if (g < N) out[g * 16 + m] = acc[r] + bias;
  }
}

// ---------------------------------------------------------------------------
extern "C" void kernel_launch(void* const* d_in, const int* in_sizes, int n_in,
                              void* d_out, int out_size, void* d_ws, size_t ws_size,
                              hipStream_t stream) {
  const float* feat   = (const float*)d_in[0];
  const int*   esrc   = (const int*)  d_in[1];
  const int*   edst   = (const int*)  d_in[2];
  const float* fc1_w  = (const float*)d_in[3];
  const float* fc1_b  = (const float*)d_in[4];
  const float* filt_w = (const float*)d_in[5];
  const float* filt_b = (const float*)d_in[6];
  const float* att_w1 = (const float*)d_in[7];
  const float* att_b1 = (const float*)d_in[8];
  const float* att_w2 = (const float*)d_in[9];
  const float* fc2_w  = (const float*)d_in[10];
  const float* fc2_b  = (const float*)d_in[11];
  float* out = (float*)d_out;

  const int N = in_sizes[0] / IN_F;
  const int E = in_sizes[1];

  float* ws   = (float*)d_ws;
  float* h1   = ws;                          // N*64
  float* xb   = ws + (size_t)N * 64;         // N*64
  float* h2   = ws + (size_t)2 * N * 64;     // N*64  (zeroed)
  float* deg  = ws + (size_t)3 * N * 64;     // deg_out N, deg_in N (zeroed)
  float* sumW = deg + (size_t)2 * N;         // 2 (zeroed)
  float* beta = sumW + 2;                    // 2

  // zero h2 + degrees + sumW (contiguous region)
  long ztot = (long)N * 64 + 2L * N + 2;
  init_ws_k<<<2048, 256, 0, stream>>>(h2, ztot);

  degree_k<<<(E + 255) / 256, 256, 0, stream>>>(esrc, edst, deg, deg + N, E);
  invsqrt_k<<<(2 * N + 255) / 256, 256, 0, stream>>>(deg, 2 * N);

  gemm_h1x_k<<<(N + 127) / 128, 256, 0, stream>>>(feat, fc1_w, fc1_b,
                                                  filt_w, filt_b, h1, xb, N);

  int scatter_threads = E * 64;
  scatter_k<<<(scatter_threads + 255) / 256, 256, 0, stream>>>(
      esrc, edst, deg, deg + N, xb, h2, E);

  att_k<<<(2 * N + 7) / 8, 256, 0, stream>>>(h1, h2, att_w1, att_b1, att_w2, sumW, N);
  beta_k<<<1, 32, 0, stream>>>(sumW, beta, N);

  out_k<<<(N + 127) / 128, 256, 0, stream>>>(h1, h2, beta, fc2_w, fc2_b, out, N);
}